// AnomalyGenerator_58944131170761
// MI455X (gfx1250) — compile-verified
//
#include <hip/hip_runtime.h>
#include <hip/hip_bf16.h>

typedef __attribute__((ext_vector_type(2))) float v2f;
typedef __attribute__((ext_vector_type(8))) float v8f;

#define KCOMP 64
#define IMG   1024
#define NPIX  (IMG * IMG)

// Workspace float layout:
//   [0    .. 255 ]  P0[k] = {mu_x, mu_y, inv_sigma_x, inv_sigma_y}   (float4 x 64)
//   [256  .. 511 ]  P1[k] = {a*log2e, c*log2e, pi, 0}                (float4 x 64)
//   [512  .. 1535]  W[k][16] : col0=pi, col1..3=pi*color_c, rest 0
//   [1536]          scale (sigmoid)
//   [1537]          global max of gmm_prob, as uint bits (init 0)

__global__ void gmm_params_kernel(const float* __restrict__ color_p,
                                  const float* __restrict__ mux_p,
                                  const float* __restrict__ muy_p,
                                  const float* __restrict__ sx_p,
                                  const float* __restrict__ sy_p,
                                  const float* __restrict__ rho_p,
                                  const float* __restrict__ pi_p,
                                  const float* __restrict__ scale_p,
                                  float* __restrict__ ws) {
    int k = threadIdx.x;  // 0..63

    // softmax over pi (K=64, each thread redundantly loops -- trivial cost)
    float pm = -3.0e38f;
    for (int j = 0; j < KCOMP; ++j) pm = fmaxf(pm, pi_p[j]);
    float psum = 0.0f;
    for (int j = 0; j < KCOMP; ++j) psum += expf(pi_p[j] - pm);
    float pik = expf(pi_p[k] - pm) / psum;

    float mux = tanhf(mux_p[k]);
    float muy = tanhf(muy_p[k]);
    float sxv = sx_p[k], syv = sy_p[k];
    float spx = (sxv > 20.0f) ? sxv : log1pf(expf(sxv));  // softplus
    float spy = (syv > 20.0f) ? syv : log1pf(expf(syv));
    float inv_sx = 1.0f / (spx + 0.01f);
    float inv_sy = 1.0f / (spy + 0.01f);
    float rho = tanhf(rho_p[k]) * 0.99f;
    float om  = 1.0f - rho * rho;
    const float L2E = 1.4426950408889634f;
    float a = L2E / (2.0f * om);      // multiplies -(zx^2+zy^2)
    float c = rho * L2E / om;         // multiplies  zx*zy

    float4* P0 = (float4*)ws;
    float4* P1 = (float4*)(ws + 256);
    float*  W  = ws + 512;
    P0[k] = make_float4(mux, muy, inv_sx, inv_sy);
    P1[k] = make_float4(a, c, pik, 0.0f);
    W[k * 16 + 0] = pik;
    W[k * 16 + 1] = pik * tanhf(color_p[0 * KCOMP + k]);
    W[k * 16 + 2] = pik * tanhf(color_p[1 * KCOMP + k]);
    W[k * 16 + 3] = pik * tanhf(color_p[2 * KCOMP + k]);
    #pragma unroll
    for (int j = 4; j < 16; ++j) W[k * 16 + j] = 0.0f;

    if (k == 0) {
        ws[1536] = 1.0f / (1.0f + expf(-scale_p[0]));
        ((unsigned int*)ws)[1537] = 0u;   // reset max each launch (determinism)
    }
}

// Pass 1: per 16-pixel tile (one wave), compute F[16x64] then D = F @ W via
// sixteen v_wmma_f32_16x16x4_f32 accumulations. Store unnormalized partials
// (S = sum f*pi into the mask plane, P_c = sum f*pi*color_c into channel
// planes) directly into d_out; reduce global max of S.
__global__ __launch_bounds__(256) void gmm_pass1_kernel(float* __restrict__ ws,
                                                        float* __restrict__ out) {
    __shared__ float4 sP0[KCOMP];
    __shared__ float4 sP1[KCOMP];
    __shared__ __align__(16) float sW[KCOMP * 16];
    __shared__ unsigned int sMax;

    int t = threadIdx.x;
    if (t < KCOMP) {
        sP0[t] = ((const float4*)ws)[t];
        sP1[t] = ((const float4*)(ws + 256))[t];
    }
    ((float4*)sW)[t] = ((const float4*)(ws + 512))[t];  // 256 * float4 = 1024 floats
    if (t == 0) sMax = 0u;
    __syncthreads();

    int wave = t >> 5;
    int lane = t & 31;
    int hh   = lane >> 4;   // which half of the wave (K pair / row group)
    int mm   = lane & 15;   // A: pixel row M   B/D: column N

    int y      = blockIdx.y;
    int x_base = (blockIdx.x * 8 + wave) * 16;

    const float S2 = 2.0f / (float)(IMG - 1);
    float xn = (float)(x_base + mm) * S2 - 1.0f;
    float yn = (float)y * S2 - 1.0f;

    v8f acc = {};
    #pragma unroll
    for (int chunk = 0; chunk < 16; ++chunk) {
        int k0 = chunk * 4 + hh * 2;
        float4 p0a = sP0[k0],     p1a = sP1[k0];
        float4 p0b = sP0[k0 + 1], p1b = sP1[k0 + 1];

        float zx = (xn - p0a.x) * p0a.z;
        float zy = (yn - p0a.y) * p0a.w;
        float tq = fmaf(zx, zx, zy * zy);
        float a0 = __builtin_amdgcn_exp2f(fmaf(zx * zy, p1a.y, -(tq * p1a.x)));

        zx = (xn - p0b.x) * p0b.z;
        zy = (yn - p0b.y) * p0b.w;
        tq = fmaf(zx, zx, zy * zy);
        float a1 = __builtin_amdgcn_exp2f(fmaf(zx * zy, p1b.y, -(tq * p1b.x)));

        v2f A; A.x = a0; A.y = a1;                        // A[M=mm][K=2h,2h+1]
        v2f B; B.x = sW[k0 * 16 + mm];                    // B[K=2h  ][N=mm]
               B.y = sW[(k0 + 1) * 16 + mm];              // B[K=2h+1][N=mm]
        acc = __builtin_amdgcn_wmma_f32_16x16x4_f32(
                  false, A, false, B, (short)0, acc, false, false);
    }

    // D layout: lane<16 holds rows M=0..7 (VGPR j -> row j), lane>=16 rows 8..15.
    // Column = mm. Columns 0..3 carry {S, P0, P1, P2}.
    if (mm < 4) {
        size_t plane = (mm == 0) ? (size_t)3 * NPIX : (size_t)(mm - 1) * NPIX;
        size_t base  = plane + (size_t)y * IMG + (size_t)x_base + (size_t)(8 * hh);
        float4 lo = make_float4(acc[0], acc[1], acc[2], acc[3]);
        float4 hi = make_float4(acc[4], acc[5], acc[6], acc[7]);
        *(float4*)(out + base)     = lo;
        *(float4*)(out + base + 4) = hi;
    }
    if (mm == 0) {  // lanes 0 and 16: column 0 = S (>= 0, so uint bits are monotone)
        float mx = fmaxf(fmaxf(fmaxf(acc[0], acc[1]), fmaxf(acc[2], acc[3])),
                         fmaxf(fmaxf(acc[4], acc[5]), fmaxf(acc[6], acc[7])));
        atomicMax(&sMax, __float_as_uint(mx));
    }
    __syncthreads();
    if (t == 0) atomicMax(((unsigned int*)ws) + 1537, sMax);
}

// Pass 2: elementwise in-place fixup of d_out using global max + scale.
__global__ __launch_bounds__(256) void gmm_pass2_kernel(const float* __restrict__ imageN,
                                                        const float* __restrict__ ws,
                                                        float* __restrict__ out) {
    int i = blockIdx.x * blockDim.x + threadIdx.x;  // float4 index over NPIX/4
    const int Q = NPIX / 4;
    float maxS  = __uint_as_float(((const unsigned int*)ws)[1537]);
    float scale = ws[1536];
    float inv   = scale / maxS;

    const float4* in4  = (const float4*)imageN;
    float4*       out4 = (float4*)out;

    float4 S = out4[3 * Q + i];
    float4 sumn = make_float4(S.x * inv, S.y * inv, S.z * inv, S.w * inv);
    out4[3 * Q + i] = sumn;  // 1 - mask

    #pragma unroll
    for (int c = 0; c < 3; ++c) {
        float4 P = out4[c * Q + i];
        float4 I = in4[c * Q + i];
        float4 r;
        r.x = fmaf(I.x, 1.0f - sumn.x, P.x * inv);
        r.y = fmaf(I.y, 1.0f - sumn.y, P.y * inv);
        r.z = fmaf(I.z, 1.0f - sumn.z, P.z * inv);
        r.w = fmaf(I.w, 1.0f - sumn.w, P.w * inv);
        out4[c * Q + i] = r;
    }
}

extern "C" void kernel_launch(void* const* d_in, const int* in_sizes, int n_in,
                              void* d_out, int out_size, void* d_ws, size_t ws_size,
                              hipStream_t stream) {
    const float* imageN  = (const float*)d_in[0];
    const float* color_p = (const float*)d_in[1];
    const float* mux_p   = (const float*)d_in[2];
    const float* muy_p   = (const float*)d_in[3];
    const float* sx_p    = (const float*)d_in[4];
    const float* sy_p    = (const float*)d_in[5];
    const float* rho_p   = (const float*)d_in[6];
    const float* pi_p    = (const float*)d_in[7];
    const float* scale_p = (const float*)d_in[8];
    float* ws  = (float*)d_ws;
    float* out = (float*)d_out;

    gmm_params_kernel<<<1, KCOMP, 0, stream>>>(color_p, mux_p, muy_p, sx_p, sy_p,
                                               rho_p, pi_p, scale_p, ws);
    gmm_pass1_kernel<<<dim3(8, IMG), 256, 0, stream>>>(ws, out);
    gmm_pass2_kernel<<<NPIX / 4 / 256, 256, 0, stream>>>(imageN, ws, out);
}